// FocalLoss_16655883174015
// MI455X (gfx1250) — compile-verified
//
#include <hip/hip_runtime.h>
#include <stdint.h>

// ---------------------------------------------------------------------------
// RetinaNet focal loss for MI455X (gfx1250).
// Dominant cost: streaming 125.7 MB of classifications once -> ~5.4us at
// 23.3 TB/s. Everything else (IoU over 12.6M anchor-box pairs, reg loss) is
// tiny. So: coalesced float4 NT loads in the streaming kernel, per-anchor
// state table in scratch (L2-resident), WMMA-based wave reductions, async
// global->LDS staging of the annotation table.
// ---------------------------------------------------------------------------

typedef float f4  __attribute__((ext_vector_type(4)));
typedef float v2f __attribute__((ext_vector_type(2)));
typedef float v8f __attribute__((ext_vector_type(8)));

#if defined(__gfx1250__) && __has_builtin(__builtin_amdgcn_wmma_f32_16x16x4_f32)
#define HAVE_WMMA_F32X4 1
#else
#define HAVE_WMMA_F32X4 0
#endif

// Exact-f32 wave32 sum. WMMA path: D = A x ones, A rows = {v_m,0,v_{m+16},0}
// => every D element of row m is v_m + v_{m+16}; lane-local sum of the 8
// C/D VGPRs gives each half-wave's 16-value sum; a second ones-WMMA merges
// the two halves. All f32 RNE per ISA 7.12 -> numerically a normal tree sum.
__device__ __forceinline__ float wave_sum(float v) {
#if HAVE_WMMA_F32X4
  v2f a; a[0] = v;    a[1] = 0.0f;
  v2f o; o[0] = 1.0f; o[1] = 1.0f;          // all-ones B (layout independent)
  v8f c = {0.f, 0.f, 0.f, 0.f, 0.f, 0.f, 0.f, 0.f};
  v8f d = __builtin_amdgcn_wmma_f32_16x16x4_f32(false, a, false, o,
                                                (short)0, c, false, false);
  float s = ((d[0] + d[1]) + (d[2] + d[3])) + ((d[4] + d[5]) + (d[6] + d[7]));
  v2f a2; a2[0] = s; a2[1] = 0.0f;
  v8f d2 = __builtin_amdgcn_wmma_f32_16x16x4_f32(false, a2, false, o,
                                                 (short)0, c, false, false);
  return d2[0];
#else
  for (int off = 16; off > 0; off >>= 1) v += __shfl_xor(v, off, 32);
  return v;
#endif
}

// state encoding per anchor: >=0 -> positive with that class label
//                            -1  -> negative (targets all 0)
//                            -2  -> ignored  (targets all -1, zero loss)

// ---------------------------------------------------------------------------
// Kernel A: per-anchor IoU max/argmax, state, regression loss, n_pos.
// grid = (ceil(A/256), B), block = 256 (8 waves).
// ---------------------------------------------------------------------------
__global__ __launch_bounds__(256) void fl_anchor_kernel(
    const f4* __restrict__ anchors4,   // (A,4) [y1,x1,y2,x2]
    const f4* __restrict__ reg4,       // (B,A,4)
    const float* __restrict__ ann,     // (B,M,5) [x1,y1,x2,y2,label]
    int* __restrict__ state,           // (B,A)
    float* __restrict__ accum,         // [0..7]=cls [8..15]=reg [16..23]=npos
    int A, int M) {
  __shared__ float sAnn[160];          // M<=32 boxes * 5
  __shared__ float sred[16];
  const int tid = threadIdx.x;
  const int b   = blockIdx.y;
  const int a   = blockIdx.x * blockDim.x + tid;
  const int annN = M * 5;

  // Stage this image's annotations into LDS via async global->LDS (ASYNCcnt).
  {
    const float* g = ann + (size_t)b * annN;
    if (tid < annN) {
#if defined(__gfx1250__)
      unsigned lds_addr = (unsigned)(size_t)(void*)(&sAnn[tid]); // AS3 offset
      unsigned long long ga = (unsigned long long)(uintptr_t)(g + tid);
      asm volatile("global_load_async_to_lds_b32 %0, %1, off"
                   :: "v"(lds_addr), "v"(ga) : "memory");
#else
      sAnn[tid] = g[tid];
#endif
    }
#if defined(__gfx1250__)
#if __has_builtin(__builtin_amdgcn_s_wait_asynccnt)
    __builtin_amdgcn_s_wait_asynccnt(0);
#else
    asm volatile("s_wait_asynccnt 0x0" ::: "memory");
#endif
#endif
    __syncthreads();
  }

  float regContrib = 0.0f;
  float posContrib = 0.0f;
  if (a < A) {
    f4 an = anchors4[a];                       // [y1,x1,y2,x2]
    float ah  = an[2] - an[0];
    float aw  = an[3] - an[1];
    float acx = an[1] + 0.5f * aw;
    float acy = an[0] + 0.5f * ah;
    float best = -3.402823466e+38f;
    int   arg  = 0;
    for (int j = 0; j < M; ++j) {
      const float* bx = &sAnn[j * 5];
      float b0 = bx[0], b1 = bx[1], b2 = bx[2], b3 = bx[3], lab = bx[4];
      float iw = fmaxf(fminf(an[3], b2) - fmaxf(an[1], b0), 0.0f);
      float ih = fmaxf(fminf(an[2], b3) - fmaxf(an[0], b1), 0.0f);
      float inter = iw * ih;
      float area  = (b2 - b0) * (b3 - b1);
      float ua    = fmaxf(ah * aw + area - inter, 1e-8f);
      float iou   = inter / ua;
      float val   = (lab != -1.0f) ? iou : -1.0f;
      if (val > best) { best = val; arg = j; }   // first-max like jnp.argmax
    }
    bool pos = best >= 0.5f;
    bool neg = best < 0.4f;
    const float* gx = &sAnn[arg * 5];
    int st = pos ? (int)gx[4] : (neg ? -1 : -2);
    if (pos) {
      posContrib = 1.0f;
      float gw  = gx[2] - gx[0];
      float gh  = gx[3] - gx[1];
      float gcx = gx[0] + 0.5f * gw;
      float gcy = gx[1] + 0.5f * gh;
      gw = fmaxf(gw, 1.0f);
      gh = fmaxf(gh, 1.0f);
      f4 r = reg4[(size_t)b * A + a];
      float t0 = (gcy - acy) / ah;
      float t1 = (gcx - acx) / aw;
      float t2 = logf(gh / ah);
      float t3 = logf(gw / aw);
      float d, s = 0.0f;
      d = fabsf(t0 - r[0]); s += (d <= (1.0f/9.0f)) ? 4.5f*d*d : d - (0.5f/9.0f);
      d = fabsf(t1 - r[1]); s += (d <= (1.0f/9.0f)) ? 4.5f*d*d : d - (0.5f/9.0f);
      d = fabsf(t2 - r[2]); s += (d <= (1.0f/9.0f)) ? 4.5f*d*d : d - (0.5f/9.0f);
      d = fabsf(t3 - r[3]); s += (d <= (1.0f/9.0f)) ? 4.5f*d*d : d - (0.5f/9.0f);
      regContrib = s;
    }
    state[(size_t)b * A + a] = st;
  }

  // Block reduction (8 waves); wave part via WMMA (EXEC all-ones here).
  float rw = wave_sum(regContrib);
  float pw = wave_sum(posContrib);
  int lane = tid & 31, wid = tid >> 5;
  if (lane == 0) { sred[wid] = rw; sred[8 + wid] = pw; }
  __syncthreads();
  if (tid == 0) {
    float rs = 0.0f, ps = 0.0f;
    for (int i = 0; i < 8; ++i) { rs += sred[i]; ps += sred[8 + i]; }
    atomicAdd(&accum[8 + b], rs);
    atomicAdd(&accum[16 + b], ps);
  }
}

// ---------------------------------------------------------------------------
// Kernel B: streaming focal classification loss. grid = (1024, B), 256 thr.
// Fully coalesced float4 NT loads of the 125.7 MB class tensor.
// ---------------------------------------------------------------------------
__global__ __launch_bounds__(256) void fl_cls_kernel(
    const f4* __restrict__ cls4,       // (B, A*C/4)
    const int* __restrict__ state,     // (B, A)
    float* __restrict__ accum,
    int A, int C4) {                   // C4 = C/4
  __shared__ float sred[8];
  const int b = blockIdx.y;
  const unsigned n4 = (unsigned)A * (unsigned)C4;
  const f4* base = cls4 + (size_t)b * n4;
  const int* stb = state + (size_t)b * A;
  float acc = 0.0f;
  const unsigned stride = gridDim.x * blockDim.x;
  for (unsigned i = blockIdx.x * blockDim.x + threadIdx.x; i < n4; i += stride) {
    f4 c = __builtin_nontemporal_load(base + i);   // 126MB stream: bypass RT
    unsigned anchor = i / (unsigned)C4;
    int c0 = (int)(i - anchor * (unsigned)C4) * 4;
    int s = stb[anchor];                            // L2-resident, broadcast
    if (s != -2) {                                  // -2: ignored anchor
      #pragma unroll
      for (int k = 0; k < 4; ++k) {
        float cv = fminf(fmaxf(c[k], 1e-4f), 1.0f - 1e-4f);
        if (s == c0 + k) {
          float om = 1.0f - cv;
          acc += 0.25f * om * om * (-logf(cv));      // target==1
        } else {
          acc += 0.75f * cv * cv * (-logf(1.0f - cv)); // target==0
        }
      }
    }
  }
  float w = wave_sum(acc);
  int lane = threadIdx.x & 31, wid = threadIdx.x >> 5;
  if (lane == 0) sred[wid] = w;
  __syncthreads();
  if (threadIdx.x == 0) {
    float s = 0.0f;
    for (int i = 0; i < 8; ++i) s += sred[i];
    atomicAdd(&accum[b], s);
  }
}

// ---------------------------------------------------------------------------
// Init + finalize
// ---------------------------------------------------------------------------
__global__ void fl_zero_kernel(float* __restrict__ accum) {
  if (threadIdx.x < 24) accum[threadIdx.x] = 0.0f;
}

__global__ void fl_final_kernel(const float* __restrict__ ann,
                                const float* __restrict__ accum,
                                float* __restrict__ out, int B, int M) {
  __shared__ float scl[32], srg[32];
  int t = threadIdx.x;
  float cl = 0.0f, rg = 0.0f;
  if (t < B) {
    bool hv = false;
    for (int m = 0; m < M; ++m)
      hv = hv || (ann[(size_t)t * M * 5 + m * 5 + 4] != -1.0f);
    float np = accum[16 + t];
    float cv = accum[t] / fmaxf(np, 1.0f);
    cl = hv ? cv : 0.0f;
    rg = (np > 0.0f) ? accum[8 + t] / fmaxf(np * 4.0f, 1.0f) : 0.0f;
  }
  scl[t] = cl; srg[t] = rg;
  __syncthreads();
  if (t == 0) {
    float sc = 0.0f, sr = 0.0f;
    for (int i = 0; i < B; ++i) { sc += scl[i]; sr += srg[i]; }
    out[0] = sc / (float)B;
    out[1] = sr / (float)B;
  }
}

extern "C" void kernel_launch(void* const* d_in, const int* in_sizes, int n_in,
                              void* d_out, int out_size, void* d_ws, size_t ws_size,
                              hipStream_t stream) {
  const float* cls = (const float*)d_in[0];  // (B,A,C)
  const float* reg = (const float*)d_in[1];  // (B,A,4)
  const float* anc = (const float*)d_in[2];  // (1,A,4)
  const float* ann = (const float*)d_in[3];  // (B,M,5)

  const int A = in_sizes[2] / 4;
  const int B = in_sizes[1] / (A * 4);
  const int C = in_sizes[0] / (B * A);
  const int M = in_sizes[3] / (B * 5);

  int*   state = (int*)d_ws;                                   // B*A ints
  float* accum = (float*)((char*)d_ws + (size_t)B * A * sizeof(int)); // 24 f32

  fl_zero_kernel<<<1, 32, 0, stream>>>(accum);

  dim3 ga((A + 255) / 256, B);
  fl_anchor_kernel<<<ga, 256, 0, stream>>>((const f4*)anc, (const f4*)reg,
                                           ann, state, accum, A, M);

  dim3 gb(1024, B);
  fl_cls_kernel<<<gb, 256, 0, stream>>>((const f4*)cls, state, accum, A, C / 4);

  fl_final_kernel<<<1, 32, 0, stream>>>(ann, accum, (float*)d_out, B, M);
}